// MoEConnectionProcessor_39427799777329
// MI455X (gfx1250) — compile-verified
//
#include <hip/hip_runtime.h>
#include <math.h>

typedef __attribute__((ext_vector_type(2))) float v2f;
typedef __attribute__((ext_vector_type(4))) float v4f;
typedef __attribute__((ext_vector_type(8))) float v8f;

#define NB_K 26
#define DT_C 0.3f

// ---- LDS layout (float offsets) ----
#define OFF_WL   0        // 64x32
#define OFF_WM   2048     // 32x32
#define OFF_WU   3072     // 64x32
#define OFF_WC   5120     // 32x32
#define OFF_WD   6144     // 64x32
#define OFF_GW1  8192     // 64x16 (cols 8..15 zero-padded)
#define OFF_B    9216     // biases etc.
#define B_BL   0
#define B_BM   32
#define B_BU   64
#define B_BC   96
#define B_BD   128
#define B_LNG  160
#define B_LNB  192
#define B_GB1  224        // 8
#define B_GW2  232        // 8x3
#define B_GB2  256        // 3
#define OFF_WAVE 9504
#define W_XBUF 0          // 16x32 x tile
#define W_A    512        // A-operand staging (locagg / agg / h / xt)
#define W_B    1024       // nb staging, later dist_agg
#define W_CTX  1536       // 16x64 gate context
#define W_HG   2560       // 16x8 gate hidden
#define W_GATE 2688       // 16x4 gate probs
#define WAVE_F 2752
#define LDS_F  (OFF_WAVE + 2 * WAVE_F)   // 15008 floats = 60032 B

#define V8Z {0.f,0.f,0.f,0.f,0.f,0.f,0.f,0.f}
#define V4Z {0.f,0.f,0.f,0.f}

__device__ __forceinline__ v8f wmma4(v2f a, v2f b, v8f c) {
  // V_WMMA_F32_16X16X4_F32: D = A(16x4) x B(4x16) + C, exact fp32
  return __builtin_amdgcn_wmma_f32_16x16x4_f32(false, a, false, b, (short)0, c, false, false);
}

// Branch-free tanh on the transcendental pipe: v_exp_f32 + v_rcp_f32.
// exp argument is always <= 0 so it never overflows; saturates to +/-1.
__device__ __forceinline__ float fast_tanh(float x) {
  float ax = fabsf(x);
  float e  = __expf(-2.0f * ax);                       // v_exp_f32 (scaled)
  float r  = (1.0f - e) * __builtin_amdgcn_rcpf(1.0f + e);  // v_rcp_f32
  return copysignf(r, x);
}

__device__ __forceinline__ float gelu_tanh(float x) {
  float x3 = x * x * x;
  return 0.5f * x * (1.0f + fast_tanh(0.7978845608028654f * (x + 0.044715f * x3)));
}

// 16xN GEMM tile: A is 16 x (4*KSTEPS) row-major in LDS (stride astr),
// W is (4*KSTEPS) x {16 or 32} row-major in LDS (stride wstr).
template <int KSTEPS, bool TWO>
__device__ __forceinline__ void gemm_tile(const float* A, int astr, const float* Wt, int wstr,
                                          int lane, v8f& c0, v8f& c1) {
  const int half = lane >> 4;     // upper half-wave covers K+2 / K+3
  const int arow = lane & 15;     // A row (M)
  const int ncol = lane & 15;     // B col (N) within tile
#pragma unroll
  for (int k = 0; k < KSTEPS; ++k) {
    const int kk = k * 4 + half * 2;
    v2f a; a.x = A[arow * astr + kk]; a.y = A[arow * astr + kk + 1];
    v2f b0; b0.x = Wt[kk * wstr + ncol]; b0.y = Wt[(kk + 1) * wstr + ncol];
    c0 = wmma4(a, b0, c0);
    if (TWO) {
      v2f b1; b1.x = Wt[kk * wstr + 16 + ncol]; b1.y = Wt[(kk + 1) * wstr + 16 + ncol];
      c1 = wmma4(a, b1, c1);
    }
  }
}

// Preload all B fragments of a 32x32 weight matrix into registers (32 VGPRs),
// for GEMMs whose weight is reused many times (Wm x14, Wd_top x3).
__device__ __forceinline__ void load_bfrags32(const float* Wt, int lane, v2f (&bf)[8][2]) {
  const int half = lane >> 4;
  const int ncol = lane & 15;
#pragma unroll
  for (int k = 0; k < 8; ++k) {
    const int kk = k * 4 + half * 2;
    bf[k][0].x = Wt[kk * 32 + ncol];      bf[k][0].y = Wt[(kk + 1) * 32 + ncol];
    bf[k][1].x = Wt[kk * 32 + 16 + ncol]; bf[k][1].y = Wt[(kk + 1) * 32 + 16 + ncol];
  }
}

__device__ __forceinline__ void gemm_preB(const float* A, int astr, const v2f (&bf)[8][2],
                                          int lane, v8f& c0, v8f& c1) {
  const int half = lane >> 4;
  const int arow = lane & 15;
#pragma unroll
  for (int k = 0; k < 8; ++k) {
    const int kk = k * 4 + half * 2;
    v2f a; a.x = A[arow * astr + kk]; a.y = A[arow * astr + kk + 1];
    c0 = wmma4(a, bf[k][0], c0);
    c1 = wmma4(a, bf[k][1], c1);
  }
}

__global__ void __launch_bounds__(64)
moe_conn_wmma_kernel(const float* __restrict__ x, const float* __restrict__ nb,
                     const float* __restrict__ Wl, const float* __restrict__ bl,
                     const float* __restrict__ Wm, const float* __restrict__ bm,
                     const float* __restrict__ Wu, const float* __restrict__ bu,
                     const float* __restrict__ Wc, const float* __restrict__ bc,
                     const float* __restrict__ Wd, const float* __restrict__ bd,
                     const float* __restrict__ lng, const float* __restrict__ lnb,
                     const float* __restrict__ gW1, const float* __restrict__ gb1,
                     const float* __restrict__ gW2, const float* __restrict__ gb2,
                     float* __restrict__ outc, float* __restrict__ outg) {
  __shared__ float sm[LDS_F];
  const int tid = threadIdx.x;

  // ---- stage weights into LDS (whole block cooperates) ----
  for (int i = tid; i < 2048; i += 64) {
    sm[OFF_WL + i] = Wl[i];
    sm[OFF_WU + i] = Wu[i];
    sm[OFF_WD + i] = Wd[i];
  }
  for (int i = tid; i < 1024; i += 64) {
    sm[OFF_WM + i] = Wm[i];
    sm[OFF_WC + i] = Wc[i];
    int r = i >> 4, c = i & 15;
    sm[OFF_GW1 + i] = (c < 8) ? gW1[r * 8 + c] : 0.0f;   // zero-pad N to 16
  }
  if (tid < 32) {
    sm[OFF_B + B_BL + tid] = bl[tid];
    sm[OFF_B + B_BM + tid] = bm[tid];
    sm[OFF_B + B_BU + tid] = bu[tid];
    sm[OFF_B + B_BC + tid] = bc[tid];
    sm[OFF_B + B_BD + tid] = bd[tid];
    sm[OFF_B + B_LNG + tid] = lng[tid];
    sm[OFF_B + B_LNB + tid] = lnb[tid];
  }
  if (tid >= 32 && tid < 40) sm[OFF_B + B_GB1 + tid - 32] = gb1[tid - 32];
  if (tid >= 40 && tid < 64) sm[OFF_B + B_GW2 + tid - 40] = gW2[tid - 40];
  if (tid < 3)               sm[OFF_B + B_GB2 + tid]      = gb2[tid];
  __syncthreads();

  const int wave = tid >> 5;
  const int lane = tid & 31;
  const int row0 = (blockIdx.x * 2 + wave) * 16;
  float* pw   = sm + OFF_WAVE + wave * WAVE_F;
  float* xbuf = pw + W_XBUF;
  float* bufA = pw + W_A;
  float* bufB = pw + W_B;
  float* ctx  = pw + W_CTX;
  float* hg   = pw + W_HG;
  float* gbuf = pw + W_GATE;
  const float* Bs = sm + OFF_B;

  const int mb = (lane >> 4) * 8;   // C/D layout: VGPR r -> row mb+r
  const int n  = lane & 15;         // C/D layout: col (per N-tile)

  // ---- load x tile (16x32, fully contiguous) ----
#pragma unroll
  for (int i = 0; i < 4; ++i) {
    int f = i * 32 + lane;
    v4f v = *(const v4f*)(x + (size_t)row0 * 32 + (size_t)f * 4);
    *(v4f*)(xbuf + f * 4) = v;
  }

  // ---- preload Wm B-fragments (reused 14x per tile) ----
  v2f bfm[8][2];
  load_bfrags32(sm + OFF_WM, lane, bfm);

  // ---- stream neighbors once: sums in regs, func neighbors through WMMA ----
  v4f locs[4] = {V4Z, V4Z, V4Z, V4Z};
  v4f dists[4] = {V4Z, V4Z, V4Z, V4Z};
  v4f acts[4] = {V4Z, V4Z, V4Z, V4Z};
  v8f msg0 = V8Z, msg1 = V8Z;
  const float bm0 = Bs[B_BM + n], bm1 = Bs[B_BM + 16 + n];

  for (int j = 0; j < NB_K; ++j) {
    v4f t[4];
#pragma unroll
    for (int i = 0; i < 4; ++i) {
      int f = i * 32 + lane;               // 8 lanes per row -> coalesced b128
      int r = f >> 3, cv = f & 7;
      t[i] = *(const v4f*)(nb + ((size_t)(row0 + r) * NB_K + j) * 32 + (size_t)cv * 4);
    }
    if (j + 1 < NB_K && lane < 16) {       // global_prefetch_b8 next neighbor tile
      __builtin_prefetch(nb + ((size_t)(row0 + lane) * NB_K + (j + 1)) * 32, 0, 0);
    }
#pragma unroll
    for (int i = 0; i < 4; ++i) {
      acts[i].x += fabsf(t[i].x); acts[i].y += fabsf(t[i].y);
      acts[i].z += fabsf(t[i].z); acts[i].w += fabsf(t[i].w);
    }
    if (j < 3) {
#pragma unroll
      for (int i = 0; i < 4; ++i) locs[i] += t[i];
    } else if (j < 17) {
      // stage to LDS, GEMM against Wm (B in regs), gelu on D-layout, accumulate mean
#pragma unroll
      for (int i = 0; i < 4; ++i) *(v4f*)(bufB + (size_t)(i * 32 + lane) * 4) = t[i];
      __builtin_amdgcn_wave_barrier();
      v8f c0 = V8Z, c1 = V8Z;
      gemm_preB(bufB, 32, bfm, lane, c0, c1);
#pragma unroll
      for (int r = 0; r < 8; ++r) {
        msg0[r] += gelu_tanh(c0[r] + bm0);
        msg1[r] += gelu_tanh(c1[r] + bm1);
      }
      __builtin_amdgcn_wave_barrier();
    } else {
#pragma unroll
      for (int i = 0; i < 4; ++i) dists[i] += t[i];
    }
  }

  // ---- dump aggregates: loc->bufA, dist->bufB, activity->ctx[:,32:64] ----
  const float inv3 = (1.0f / 3.0f), inv9 = (1.0f / 9.0f);
  const float inv14 = (1.0f / 14.0f), inv26 = (1.0f / 26.0f);
#pragma unroll
  for (int i = 0; i < 4; ++i) {
    int f = i * 32 + lane;
    int r = f >> 3, cv = f & 7;
    *(v4f*)(bufA + f * 4) = locs[i] * inv3;
    *(v4f*)(bufB + f * 4) = dists[i] * inv9;
    *(v4f*)(ctx + r * 64 + 32 + cv * 4) = acts[i] * inv26;
  }
  __builtin_amdgcn_wave_barrier();

  // ---- layernorm(x) -> ctx[:,0:32] (one row per lane 0..15) ----
  if (lane < 16) {
    const float* xr = xbuf + lane * 32;
    float mu = 0.f;
    for (int i = 0; i < 32; ++i) mu += xr[i];
    mu *= (1.0f / 32.0f);
    float var = 0.f;
    for (int i = 0; i < 32; ++i) { float d = xr[i] - mu; var += d * d; }
    var *= (1.0f / 32.0f);
    float rs = rsqrtf(var + 1e-5f);
    float* cr = ctx + lane * 64;
    for (int i = 0; i < 32; ++i)
      cr[i] = (xr[i] - mu) * rs * Bs[B_LNG + i] + Bs[B_LNB + i];
  }
  __builtin_amdgcn_wave_barrier();

  // x in C/D layout
  v8f xd0, xd1;
#pragma unroll
  for (int r = 0; r < 8; ++r) {
    xd0[r] = xbuf[(mb + r) * 32 + n];
    xd1[r] = xbuf[(mb + r) * 32 + 16 + n];
  }

  // ---- branch 1: y_local = x + tanh([x, loc_agg] @ Wl + bl) ----
  v8f yl0, yl1;
  {
    v8f c0 = V8Z, c1 = V8Z;
    gemm_tile<8, true>(xbuf, 32, sm + OFF_WL, 32, lane, c0, c1);            // rows 0..31
    gemm_tile<8, true>(bufA, 32, sm + OFF_WL + 32 * 32, 32, lane, c0, c1);  // rows 32..63
    const float b0 = Bs[B_BL + n], b1 = Bs[B_BL + 16 + n];
#pragma unroll
    for (int r = 0; r < 8; ++r) {
      yl0[r] = xd0[r] + fast_tanh(c0[r] + b0);
      yl1[r] = xd1[r] + fast_tanh(c1[r] + b1);
    }
  }

  // ---- branch 2: agg -> h (Wu), y_func = h + DT*tanh(h @ Wc + bc) ----
  __builtin_amdgcn_wave_barrier();
#pragma unroll
  for (int r = 0; r < 8; ++r) {
    bufA[(mb + r) * 32 + n]      = msg0[r] * inv14;
    bufA[(mb + r) * 32 + 16 + n] = msg1[r] * inv14;
  }
  __builtin_amdgcn_wave_barrier();
  v8f h0, h1, yf0, yf1;
  {
    v8f c0 = V8Z, c1 = V8Z;
    gemm_tile<8, true>(xbuf, 32, sm + OFF_WU, 32, lane, c0, c1);
    gemm_tile<8, true>(bufA, 32, sm + OFF_WU + 32 * 32, 32, lane, c0, c1);
    const float b0 = Bs[B_BU + n], b1 = Bs[B_BU + 16 + n];
#pragma unroll
    for (int r = 0; r < 8; ++r) {
      h0[r] = xd0[r] + fast_tanh(c0[r] + b0);
      h1[r] = xd1[r] + fast_tanh(c1[r] + b1);
    }
  }
  __builtin_amdgcn_wave_barrier();
#pragma unroll
  for (int r = 0; r < 8; ++r) {
    bufA[(mb + r) * 32 + n]      = h0[r];
    bufA[(mb + r) * 32 + 16 + n] = h1[r];
  }
  __builtin_amdgcn_wave_barrier();
  {
    v8f c0 = V8Z, c1 = V8Z;
    gemm_tile<8, true>(bufA, 32, sm + OFF_WC, 32, lane, c0, c1);
    const float b0 = Bs[B_BC + n], b1 = Bs[B_BC + 16 + n];
#pragma unroll
    for (int r = 0; r < 8; ++r) {
      yf0[r] = h0[r] + DT_C * fast_tanh(c0[r] + b0);
      yf1[r] = h1[r] + DT_C * fast_tanh(c1[r] + b1);
    }
  }

  // ---- branch 3: CNF, dist-half of Wd hoisted (loop-invariant) ----
  v8f d0 = V8Z, d1 = V8Z;
  gemm_tile<8, true>(bufB, 32, sm + OFF_WD + 32 * 32, 32, lane, d0, d1);    // dist_agg @ Wd[32:64]
  v2f bfd[8][2];
  load_bfrags32(sm + OFF_WD, lane, bfd);                                    // Wd top, reused 3x
  v8f xt0 = xd0, xt1 = xd1;
  const float bd0 = Bs[B_BD + n], bd1 = Bs[B_BD + 16 + n];
  for (int s = 0; s < 3; ++s) {
    __builtin_amdgcn_wave_barrier();
#pragma unroll
    for (int r = 0; r < 8; ++r) {
      bufA[(mb + r) * 32 + n]      = xt0[r];
      bufA[(mb + r) * 32 + 16 + n] = xt1[r];
    }
    __builtin_amdgcn_wave_barrier();
    v8f c0 = d0, c1 = d1;                         // seed with constant half
    gemm_preB(bufA, 32, bfd, lane, c0, c1);
#pragma unroll
    for (int r = 0; r < 8; ++r) {
      xt0[r] += DT_C * fast_tanh(c0[r] + bd0);
      xt1[r] += DT_C * fast_tanh(c1[r] + bd1);
    }
  }

  // ---- gate: ctx(16x64) @ gW1p(64x16) via WMMA (K=64 -> 16 steps) ----
  {
    v8f g = V8Z, dummy = V8Z;
    gemm_tile<16, false>(ctx, 64, sm + OFF_GW1, 16, lane, g, dummy);
    if (n < 8) {
      const float b0 = Bs[B_GB1 + n];
#pragma unroll
      for (int r = 0; r < 8; ++r) hg[(mb + r) * 8 + n] = gelu_tanh(g[r] + b0);
    }
  }
  __builtin_amdgcn_wave_barrier();
  if (lane < 16) {   // 16x8 @ 8x3 + softmax, one row per lane
    float z0 = Bs[B_GB2 + 0], z1 = Bs[B_GB2 + 1], z2 = Bs[B_GB2 + 2];
    for (int k = 0; k < 8; ++k) {
      float hv = hg[lane * 8 + k];
      z0 += hv * Bs[B_GW2 + k * 3 + 0];
      z1 += hv * Bs[B_GW2 + k * 3 + 1];
      z2 += hv * Bs[B_GW2 + k * 3 + 2];
    }
    float m = fmaxf(z0, fmaxf(z1, z2));
    float e0 = __expf(z0 - m), e1 = __expf(z1 - m), e2 = __expf(z2 - m);
    float inv = __builtin_amdgcn_rcpf(e0 + e1 + e2);
    float p0 = e0 * inv, p1 = e1 * inv, p2 = e2 * inv;
    gbuf[lane * 4 + 0] = p0; gbuf[lane * 4 + 1] = p1; gbuf[lane * 4 + 2] = p2;
    size_t go = (size_t)(row0 + lane) * 3;
    outg[go + 0] = p0; outg[go + 1] = p1; outg[go + 2] = p2;
  }
  __builtin_amdgcn_wave_barrier();

  // ---- combine branches with gate weights, store combined ----
#pragma unroll
  for (int r = 0; r < 8; ++r) {
    int m = mb + r;
    float g0 = gbuf[m * 4 + 0], g1 = gbuf[m * 4 + 1], g2 = gbuf[m * 4 + 2];
    size_t ro = (size_t)(row0 + m) * 32;
    outc[ro + n]      = g0 * yl0[r] + g1 * yf0[r] + g2 * xt0[r];
    outc[ro + 16 + n] = g0 * yl1[r] + g1 * yf1[r] + g2 * xt1[r];
  }
}

extern "C" void kernel_launch(void* const* d_in, const int* in_sizes, int n_in,
                              void* d_out, int out_size, void* d_ws, size_t ws_size,
                              hipStream_t stream) {
  (void)n_in; (void)out_size; (void)d_ws; (void)ws_size;
  const float* x   = (const float*)d_in[0];
  const float* nbp = (const float*)d_in[1];
  const float* Wl  = (const float*)d_in[2];
  const float* bl  = (const float*)d_in[3];
  const float* Wm  = (const float*)d_in[4];
  const float* bm  = (const float*)d_in[5];
  const float* Wu  = (const float*)d_in[6];
  const float* bu  = (const float*)d_in[7];
  const float* Wc  = (const float*)d_in[8];
  const float* bc  = (const float*)d_in[9];
  const float* Wd  = (const float*)d_in[10];
  const float* bd  = (const float*)d_in[11];
  const float* lng = (const float*)d_in[12];
  const float* lnb = (const float*)d_in[13];
  const float* gW1 = (const float*)d_in[14];
  const float* gb1 = (const float*)d_in[15];
  const float* gW2 = (const float*)d_in[16];
  const float* gb2 = (const float*)d_in[17];

  int B = in_sizes[0] / 32;
  float* outc = (float*)d_out;
  float* outg = outc + (size_t)B * 32;

  int blocks = B / 32;   // 2 waves/block, 16 rows/wave
  moe_conn_wmma_kernel<<<blocks, 64, 0, stream>>>(
      x, nbp, Wl, bl, Wm, bm, Wu, bu, Wc, bc, Wd, bd, lng, lnb,
      gW1, gb1, gW2, gb2, outc, outg);
}